// Quan2d_23613730193912
// MI455X (gfx1250) — compile-verified
//
#include <hip/hip_runtime.h>

typedef float v2f __attribute__((ext_vector_type(2)));
typedef float v4f __attribute__((ext_vector_type(4)));
typedef float v8f __attribute__((ext_vector_type(8)));

#define LDS_W 36
#define LDS_H 34

// ---------------------------------------------------------------------------
// Setup kernel: build the 16x16 circuit matrix W[n][k] = U[n][k]  (y = U v).
// Thread j simulates the circuit on basis vector e_j -> column j of U.
// Wire w lives on bit (3-w) of the state index (wire 0 = MSB).
// ---------------------------------------------------------------------------
__device__ inline void ry_gate(float* s, int bp, float theta) {
    float c = cosf(0.5f * theta);
    float sn = sinf(0.5f * theta);
#pragma unroll
    for (int i = 0; i < 16; ++i) {
        if (((i >> bp) & 1) == 0) {
            int i1 = i | (1 << bp);
            float a0 = s[i], a1 = s[i1];
            s[i]  = c * a0 - sn * a1;
            s[i1] = sn * a0 + c * a1;
        }
    }
}

__device__ inline void cnot_gate(float* s, int cb, int tb) {
#pragma unroll
    for (int i = 0; i < 16; ++i) {
        if (((i >> cb) & 1) == 1 && ((i >> tb) & 1) == 0) {
            int i1 = i | (1 << tb);
            float t = s[i]; s[i] = s[i1]; s[i1] = t;
        }
    }
}

__global__ __launch_bounds__(16) void build_u_kernel(const float* __restrict__ w8,
                                                     float* __restrict__ W) {
    int j = threadIdx.x;
    if (j >= 16) return;
    float s[16];
#pragma unroll
    for (int i = 0; i < 16; ++i) s[i] = (i == j) ? 1.0f : 0.0f;
    // RY layer 1: wires 0..3 -> bits 3..0
#pragma unroll
    for (int w = 0; w < 4; ++w) ry_gate(s, 3 - w, w8[w]);
    // CNOT ring, applied sequentially: (0,1),(1,2),(2,3),(3,0)
    cnot_gate(s, 3, 2);
    cnot_gate(s, 2, 1);
    cnot_gate(s, 1, 0);
    cnot_gate(s, 0, 3);
    // RY layer 2
#pragma unroll
    for (int w = 0; w < 4; ++w) ry_gate(s, 3 - w, w8[4 + w]);
    // column j of U -> W[n*16 + j]
#pragma unroll
    for (int n = 0; n < 16; ++n) W[n * 16 + j] = s[n];
}

// One WHT butterfly stage (XOR distance M) applied to 8 values at once:
// 8 independent ds_swizzles issued back-to-back, then 8 combines, so the
// DS latency pipelines instead of serializing on s_wait_dscnt per value.
template <int M>
__device__ inline void wht_stage8(float p[8], int lane) {
    float o[8];
#pragma unroll
    for (int v = 0; v < 8; ++v)
        o[v] = __int_as_float(
            __builtin_amdgcn_ds_swizzle(__float_as_int(p[v]), 0x1f | (M << 10)));
#pragma unroll
    for (int v = 0; v < 8; ++v)
        p[v] = (lane & M) ? (o[v] - p[v]) : (p[v] + o[v]);
}

// ---------------------------------------------------------------------------
// Main kernel: one block per image. 8 waves x 2 tiles x 16 patches = 256
// patches/image. GEMM tile: A = 16 patches x 16 pixels (fp32), B = U^T,
// chained V_WMMA_F32_16X16X4_F32 over 4 K-chunks. Reduction: square +
// 16-pt Walsh-Hadamard across lanes; DC term = ||v||^2 (U orthogonal).
// ---------------------------------------------------------------------------
__global__ __launch_bounds__(256) void quanv_kernel(const float* __restrict__ x,
                                                    const float* __restrict__ W,
                                                    float* __restrict__ out) {
    __shared__ float img[LDS_H * LDS_W];

    const int b   = blockIdx.x;
    const int tid = threadIdx.x;
    const float* xb = x + (size_t)b * 1024;

    // Interior: 256 threads x float4 = 32x32 image (b128 load + b128 LDS store).
    {
        int r  = tid >> 3;          // 0..31
        int c4 = (tid & 7) << 2;    // 0,4,...,28
        v4f vals = *(const v4f*)(xb + r * 32 + c4);
        *(v4f*)(&img[r * LDS_W + c4]) = vals;
    }
    // Pad halo (disjoint from interior): rows 32..33 all cols, cols 32..35 rows 0..31.
    if (tid < 72) {
        int r = 32 + tid / 36, c = tid % 36;
        img[r * LDS_W + c] = 0.0f;
    } else if (tid < 200) {
        int u = tid - 72;
        int r = u >> 2, c = 32 + (u & 3);
        img[r * LDS_W + c] = 0.0f;
    }
    __syncthreads();

    const int lane = tid & 31;
    const int wv   = tid >> 5;     // wave id 0..7
    const int l16  = lane & 15;
    const int h    = lane >> 4;    // half-wave 0/1

    // B fragments: B_c[k'][n] = U[n][4c+k'];  VGPR0 = rows {0,2}, VGPR1 = {1,3}
    v2f bfrag[4];
#pragma unroll
    for (int c = 0; c < 4; ++c) {
        const float* p = W + l16 * 16 + 4 * c + 2 * h;
        bfrag[c] = *(const v2f*)p;
    }

    // This lane's WHT output slot: l16 = 8,4,2,1 -> wire 0,1,2,3.
    const bool active = (l16 == 8) | (l16 == 4) | (l16 == 2) | (l16 == 1);
    int wIdx = 0;
    if      (l16 == 4) wIdx = 1;
    else if (l16 == 2) wIdx = 2;
    else if (l16 == 1) wIdx = 3;

    // Per-lane output base: out[b*1024 + (pb + v + 8h)*4 + wIdx]
    //   = outb + (wIdx + 32h) + 4*pb + 4*v
    float* const po = out + (size_t)b * 1024 + wIdx + (h << 5);

#pragma unroll
    for (int t = 0; t < 2; ++t) {
        const int pb    = wv * 32 + t * 16;   // patch base of this tile
        const int patch = pb + l16;           // A-matrix row for this lane
        const int pr = patch >> 4, pc = patch & 15;

        // A fragments: chunk c = patch row c; lane half selects cols {0,1}/{2,3}
        v2f afrag[4];
#pragma unroll
        for (int c = 0; c < 4; ++c) {
            const float* p = &img[(2 * pr + c) * LDS_W + 2 * pc + 2 * h];
            afrag[c] = *(const v2f*)p;
        }

        v8f acc = {};
#pragma unroll
        for (int c = 0; c < 4; ++c) {
            acc = __builtin_amdgcn_wmma_f32_16x16x4_f32(
                /*neg_a=*/false, afrag[c],
                /*neg_b=*/false, bfrag[c],
                /*c_mod=*/(short)0, acc,
                /*reuse_a=*/false, /*reuse_b=*/false);
        }

        // Square, then 4 WHT stages applied stage-major over all 8 D-regs.
        float p[8];
#pragma unroll
        for (int v = 0; v < 8; ++v) p[v] = acc[v] * acc[v];
        wht_stage8<1>(p, lane);
        wht_stage8<2>(p, lane);
        wht_stage8<4>(p, lane);
        wht_stage8<8>(p, lane);

        // Broadcast DC term (||v||^2) from lane (h*16) to each half, all 8 at once.
        float nb[8];
#pragma unroll
        for (int v = 0; v < 8; ++v)
            nb[v] = __int_as_float(
                __builtin_amdgcn_ds_swizzle(__float_as_int(p[v]), 0x10));

        float val[8];
#pragma unroll
        for (int v = 0; v < 8; ++v) {
            float r = __builtin_amdgcn_rcpf(nb[v]);   // + 1 Newton step
            r = r * (2.0f - nb[v] * r);
            val[v] = p[v] * r;
        }

        if (active) {
            float* pt = po + pb * 4;
#pragma unroll
            for (int v = 0; v < 8; ++v) pt[v * 4] = val[v];
        }
    }
}

extern "C" void kernel_launch(void* const* d_in, const int* in_sizes, int n_in,
                              void* d_out, int out_size, void* d_ws, size_t ws_size,
                              hipStream_t stream) {
    (void)in_sizes; (void)n_in; (void)out_size; (void)ws_size;
    const float* x       = (const float*)d_in[0];   // [8192,1,32,32] fp32
    const float* weights = (const float*)d_in[1];   // [1,8] fp32
    float* out = (float*)d_out;                     // [8192,1024] fp32
    float* U   = (float*)d_ws;                      // 256 floats

    build_u_kernel<<<1, 16, 0, stream>>>(weights, U);
    quanv_kernel<<<8192, 256, 0, stream>>>(x, U, out);
}